// GraphSAGEModel_15676630631014
// MI455X (gfx1250) — compile-verified
//
#include <hip/hip_runtime.h>

typedef float v2f __attribute__((ext_vector_type(2)));
typedef float v8f __attribute__((ext_vector_type(8)));

#define D 128          // feature width (in == out)
#define TILE_M 16      // rows per block in GEMM

// Native f32 global atomic add (no return form, tracked by STOREcnt; s_endpgm
// drains it). Guarantees we never fall back to a CAS loop.
__device__ __forceinline__ void atomic_add_f32(float* p, float v) {
    asm volatile("global_atomic_add_f32 %0, %1, off"
                 :
                 : "v"(p), "v"(v)
                 : "memory");
}

// ---------------------------------------------------------------------------
// Zero a float buffer.
__global__ __launch_bounds__(256) void k_zero(float* __restrict__ p, int n) {
    int i = blockIdx.x * 256 + threadIdx.x;
    if (i < n) p[i] = 0.0f;
}

// ---------------------------------------------------------------------------
// Degree: one thread per edge, atomic add 1.0 at dst.
__global__ __launch_bounds__(256) void k_deg(const long long* __restrict__ ei,
                                             float* __restrict__ deg, int nE) {
    int e = blockIdx.x * 256 + threadIdx.x;
    if (e >= nE) return;
    long long dst = ei[nE + e];
    atomic_add_f32(&deg[dst], 1.0f);
}

// ---------------------------------------------------------------------------
// invdeg[n] = 1 / max(deg[n], 1)  (computed once, reused by both layers)
__global__ __launch_bounds__(256) void k_invdeg(float* __restrict__ deg, int n) {
    int i = blockIdx.x * 256 + threadIdx.x;
    if (i < n) deg[i] = 1.0f / fmaxf(deg[i], 1.0f);
}

// ---------------------------------------------------------------------------
// Scatter: one wave (32 lanes) per edge; each lane moves one float4 of the
// 128-float row. Row read is a contiguous 512B burst; adds are native f32
// global atomics.
__global__ __launch_bounds__(256) void k_scatter(const float* __restrict__ x,
                                                 const long long* __restrict__ ei,
                                                 float* __restrict__ agg, int nE) {
    int i = blockIdx.x * 256 + threadIdx.x;
    int e = i >> 5;          // whole wave shares one edge
    int c = i & 31;          // float4 chunk within the row
    if (e >= nE) return;
    long long src = ei[e];
    long long dst = ei[nE + e];
    float4 v = ((const float4*)(x + (size_t)src * D))[c];
    float* a = agg + (size_t)dst * D + c * 4;
    atomic_add_f32(a + 0, v.x);
    atomic_add_f32(a + 1, v.y);
    atomic_add_f32(a + 2, v.z);
    atomic_add_f32(a + 3, v.w);
}

// ---------------------------------------------------------------------------
// mean = agg * invdeg, in place. n = N*D. Pure load-mul-store (bandwidth).
__global__ __launch_bounds__(256) void k_mean(float* __restrict__ agg,
                                              const float* __restrict__ invdeg,
                                              int n) {
    int i = blockIdx.x * 256 + threadIdx.x;
    if (i >= n) return;
    agg[i] = agg[i] * invdeg[i >> 7];     // i / D
}

// ---------------------------------------------------------------------------
// Fused SAGE linear: out = relu( mean @ Wl + b + x @ Wr ), all f32.
// Block = 256 threads = 8 waves; block computes a 16x128 output stripe.
// Wave w owns 16 output columns [w*16, w*16+16). K = 128 in steps of 4 using
// V_WMMA_F32_16X16X4_F32. A-tiles (mean rows, x rows) staged in LDS.
__global__ __launch_bounds__(256) void k_sage_gemm(
    const float* __restrict__ mean,   // [N,128]
    const float* __restrict__ xin,    // [N,128]
    const float* __restrict__ Wl,     // [128,128] row-major (k, col)
    const float* __restrict__ bias,   // [128]
    const float* __restrict__ Wr,     // [128,128]
    float* __restrict__ out)          // [N,128]
{
    __shared__ float sM[TILE_M * D];
    __shared__ float sX[TILE_M * D];

    const int tid   = threadIdx.x;
    const int node0 = blockIdx.x * TILE_M;

    // Stage 16x128 tiles of mean and x: 512 float4 per tile, 256 threads.
    {
        const float4* gM = (const float4*)(mean + (size_t)node0 * D);
        const float4* gX = (const float4*)(xin  + (size_t)node0 * D);
        float4* s4M = (float4*)sM;
        float4* s4X = (float4*)sX;
        s4M[tid]       = gM[tid];
        s4M[tid + 256] = gM[tid + 256];
        s4X[tid]       = gX[tid];
        s4X[tid + 256] = gX[tid + 256];
    }
    __syncthreads();

    const int wave = tid >> 5;        // 0..7 -> 16-column slab
    const int lane = tid & 31;
    const int half = lane >> 4;       // 0: K pair {k0,k0+1}; 1: {k0+2,k0+3}
    const int l16  = lane & 15;
    const int col  = wave * 16 + l16; // output column for this lane

    // C/D layout: VGPR r <-> row (r + 8*half), col l16. Bias depends on col
    // only -> broadcast into all 8 accumulator registers.
    v8f acc;
    {
        float bv = bias[col];
#pragma unroll
        for (int i = 0; i < 8; ++i) acc[i] = bv;
    }

    // acc += mean_tile @ Wl
#pragma unroll
    for (int k0 = 0; k0 < D; k0 += 4) {
        int kb = k0 + 2 * half;
        v2f a = *(const v2f*)&sM[l16 * D + kb];           // A[l16][kb..kb+1]
        v2f b;
        b.x = Wl[(size_t)kb * D + col];                   // B[kb][col]
        b.y = Wl[(size_t)(kb + 1) * D + col];             // B[kb+1][col]
        acc = __builtin_amdgcn_wmma_f32_16x16x4_f32(
            false, a, false, b, (short)0, acc, false, false);
    }

    // acc += x_tile @ Wr
#pragma unroll
    for (int k0 = 0; k0 < D; k0 += 4) {
        int kb = k0 + 2 * half;
        v2f a = *(const v2f*)&sX[l16 * D + kb];
        v2f b;
        b.x = Wr[(size_t)kb * D + col];
        b.y = Wr[(size_t)(kb + 1) * D + col];
        acc = __builtin_amdgcn_wmma_f32_16x16x4_f32(
            false, a, false, b, (short)0, acc, false, false);
    }

    // ReLU + store. Row of acc[i] = node0 + i + 8*half.
#pragma unroll
    for (int i = 0; i < 8; ++i) {
        float v = acc[i];
        v = v > 0.0f ? v : 0.0f;
        out[(size_t)(node0 + i + 8 * half) * D + col] = v;
    }
}

// ---------------------------------------------------------------------------
extern "C" void kernel_launch(void* const* d_in, const int* in_sizes, int n_in,
                              void* d_out, int out_size, void* d_ws, size_t ws_size,
                              hipStream_t stream) {
    const float*     x   = (const float*)d_in[0];
    const long long* ei  = (const long long*)d_in[1];   // int64 [2, E]
    const float*     W1l = (const float*)d_in[2];
    const float*     b1  = (const float*)d_in[3];
    const float*     W1r = (const float*)d_in[4];
    const float*     W2l = (const float*)d_in[5];
    const float*     b2  = (const float*)d_in[6];
    const float*     W2r = (const float*)d_in[7];
    float*           out = (float*)d_out;

    const int N = in_sizes[0] / D;      // 50000
    const int E = in_sizes[1] / 2;      // 640000

    // Workspace layout (all 16B-aligned given sizes):
    float* agg = (float*)d_ws;                     // [N, D]
    float* deg = agg + (size_t)N * D;              // [N] -> becomes invdeg
    float* h1  = deg + N;                          // [N, D]

    const int nAgg = N * D;
    dim3 blk(256);
    dim3 gZeroAll((nAgg + N + 255) / 256);
    dim3 gZeroAgg((nAgg + 255) / 256);
    dim3 gDeg((E + 255) / 256);
    dim3 gNode((N + 255) / 256);
    dim3 gScat(((size_t)E * 32 + 255) / 256);
    dim3 gMean((nAgg + 255) / 256);
    dim3 gGemm(N / TILE_M);

    // --- Degree / inverse degree (graph-only, shared by both layers) -------
    k_zero<<<gZeroAll, blk, 0, stream>>>(agg, nAgg + N);   // agg and deg
    k_deg<<<gDeg, blk, 0, stream>>>(ei, deg, E);
    k_invdeg<<<gNode, blk, 0, stream>>>(deg, N);

    // --- Layer 1 -----------------------------------------------------------
    k_scatter<<<gScat, blk, 0, stream>>>(x, ei, agg, E);
    k_mean<<<gMean, blk, 0, stream>>>(agg, deg, nAgg);
    k_sage_gemm<<<gGemm, blk, 0, stream>>>(agg, x, W1l, b1, W1r, h1);

    // --- Layer 2 -----------------------------------------------------------
    k_zero<<<gZeroAgg, blk, 0, stream>>>(agg, nAgg);
    k_scatter<<<gScat, blk, 0, stream>>>(h1, ei, agg, E);
    k_mean<<<gMean, blk, 0, stream>>>(agg, deg, nAgg);
    k_sage_gemm<<<gGemm, blk, 0, stream>>>(agg, h1, W2l, b2, W2r, out);
}